// Assigner_72430328480062
// MI455X (gfx1250) — compile-verified
//
#include <hip/hip_runtime.h>
#include <hip/hip_bf16.h>
#include <stdint.h>

#define Bn 4
#define Nn 65536
#define Mn 128
#define NUM_POS 128
#define NUM_NEG 384
#define NUM_SAMPLES 512
#define EPSI 1e-6f
#define BIGF 10.0f
#define POS_THR 0.5f
#define NEG_THR 0.5f
#define MIN_POS 0.5f

// float offsets in d_out
#define OUT_ROIS 0
#define OUT_LABELS (Bn * NUM_SAMPLES * 8)
#define OUT_VALID (OUT_LABELS + Bn * NUM_SAMPLES)
#define OUT_ASSIGNED (OUT_VALID + Bn * NUM_SAMPLES)
#define OUT_MAXOV (OUT_ASSIGNED + Bn * Nn)

typedef unsigned int uint4v __attribute__((ext_vector_type(4)));
typedef int int8v __attribute__((ext_vector_type(8)));
typedef int int4v __attribute__((ext_vector_type(4)));

// ---- CDNA5-specific helpers -------------------------------------------------
// Async global->LDS copy (gfx1250 ASYNCcnt path). The low 32 bits of a generic
// pointer to __shared__ equal the wave-relative LDS byte offset (flat->LDS
// aperture truncates to addr[31:0]).
__device__ __forceinline__ void async_copy_b128(void* lds_ptr, const void* gptr) {
  unsigned lds_off = (unsigned)(unsigned long long)(lds_ptr);
  asm volatile("global_load_async_to_lds_b128 %0, %1, off"
               :: "v"(lds_off), "v"(gptr)
               : "memory");
}
__device__ __forceinline__ void wait_async0() {
  asm volatile("s_wait_asynccnt 0" ::: "memory");
}

// Tensor Data Mover: DMA a contiguous 1-D tile of `dwords` DWORDs from global
// memory to LDS.  D# built per CDNA5 ISA:
//   group0: count=1 | lds_addr | global_addr[56:0] | type=2
//   group1: data_size=2 (4B), tensor_dim0 = tile_dim0 = dwords, strides 0
//   remaining groups: zero (<=2-D tensor)
// This toolchain is the clang-23 / therock-10.0 lane -> 6-arg builtin
// (uint32x4, int32x8, int32x4, int32x4, int32x8, i32 cpol).
__device__ __forceinline__ void tdm_load_to_lds(void* lds_ptr, const void* gptr,
                                                unsigned dwords) {
  const unsigned lds_off = (unsigned)(unsigned long long)(lds_ptr);
  const unsigned long long ga = (unsigned long long)gptr;
  uint4v g0;
  g0.x = 1u;                                        // count=1, user mode
  g0.y = lds_off;                                   // lds_addr
  g0.z = (unsigned)(ga & 0xffffffffu);              // global_addr[31:0]
  g0.w = (unsigned)((ga >> 32) & 0x01ffffffu) |     // global_addr[56:32]
         (2u << 30);                                // type=2 ("image")
  int8v g1;
  g1[0] = (int)(2u << 16);                          // data_size=4B, wg_mask=0
  g1[1] = (int)((dwords & 0xffffu) << 16);          // tensor_dim0[15:0]
  g1[2] = (int)(dwords >> 16);                      // tensor_dim0[31:16]
  g1[3] = (int)((dwords & 0xffffu) << 16);          // tile_dim0[15:0]
  g1[4] = 0;                                        // tile_dim1/2 unused
  g1[5] = 0;                                        // tensor_dim0_stride
  g1[6] = 0;
  g1[7] = 0;
  int4v gz4 = {0, 0, 0, 0};
  int8v gz8 = {0, 0, 0, 0, 0, 0, 0, 0};
  __builtin_amdgcn_tensor_load_to_lds(g0, g1, gz4, gz4, gz8, 0);
}
__device__ __forceinline__ void wait_tensor0() {
  __builtin_amdgcn_s_wait_tensorcnt(0);
}

// ---- kernel 0: zero per-GT max bits ----------------------------------------
__global__ void init_gtmax_kernel(unsigned* __restrict__ gtmax) {
  int t = blockIdx.x * blockDim.x + threadIdx.x;
  if (t < Bn * Mn) gtmax[t] = 0u;
}

// ---- kernel 1: per-anchor max IoU + per-GT max (column) ---------------------
// GT tile (2KB) staged via Tensor Data Mover: one DMA issued by wave 0.
__global__ __launch_bounds__(256)
void iou_gtmax_kernel(const float* __restrict__ bboxes,
                      const float* __restrict__ gtb,
                      unsigned* __restrict__ gtmax,
                      float* __restrict__ out) {
  __shared__ float sgt[Mn * 4];
  __shared__ float sarea[Mn];
  __shared__ unsigned sgmax[Mn];
  const int tid = threadIdx.x;
  const int b = blockIdx.x >> 8;      // 256 blocks per batch
  const int chunk = blockIdx.x & 255;

  if (tid < 32) {                      // wave 0 issues the TDM DMA
    tdm_load_to_lds(sgt, gtb + (size_t)b * Mn * 4, Mn * 4);
    wait_tensor0();
  }
  if (tid < Mn) sgmax[tid] = 0u;
  __syncthreads();
  if (tid < Mn) {
    float x1 = sgt[tid*4], y1 = sgt[tid*4+1], x2 = sgt[tid*4+2], y2 = sgt[tid*4+3];
    sarea[tid] = (x2 - x1) * (y2 - y1);
  }
  __syncthreads();

  const int n = chunk * 256 + tid;
  const float4 bb = reinterpret_cast<const float4*>(bboxes)[(size_t)b * Nn + n];
  const float ab = (bb.z - bb.x) * (bb.w - bb.y);
  float best = -1.0f;
  #pragma unroll 4
  for (int m = 0; m < Mn; ++m) {
    float lx = fmaxf(bb.x, sgt[m*4+0]);
    float ly = fmaxf(bb.y, sgt[m*4+1]);
    float rx = fminf(bb.z, sgt[m*4+2]);
    float ry = fminf(bb.w, sgt[m*4+3]);
    float w = fmaxf(rx - lx, 0.0f);
    float h = fmaxf(ry - ly, 0.0f);
    float inter = w * h;
    float iou = inter / (ab + sarea[m] - inter + EPSI);
    best = fmaxf(best, iou);
    // wave32 cross-lane max, then one LDS atomic per wave per m
    float v = iou;
    v = fmaxf(v, __shfl_xor(v, 16, 32));
    v = fmaxf(v, __shfl_xor(v, 8, 32));
    v = fmaxf(v, __shfl_xor(v, 4, 32));
    v = fmaxf(v, __shfl_xor(v, 2, 32));
    v = fmaxf(v, __shfl_xor(v, 1, 32));
    if ((tid & 31) == 0) atomicMax(&sgmax[m], __float_as_uint(v)); // iou >= 0
  }
  out[OUT_MAXOV + (size_t)b * Nn + n] = best;
  __syncthreads();
  if (tid < Mn) atomicMax(&gtmax[b * Mn + tid], sgmax[tid]);
}

// ---- kernel 2: assignment (argmax + low-quality match) ----------------------
// GT tile staged via the async global->LDS path (ASYNCcnt).
__global__ __launch_bounds__(256)
void assign_kernel(const float* __restrict__ bboxes,
                   const float* __restrict__ gtb,
                   const unsigned* __restrict__ gtmax,
                   float* __restrict__ out) {
  __shared__ float sgt[Mn * 4];
  __shared__ float sarea[Mn];
  __shared__ float sgm[Mn];
  const int tid = threadIdx.x;
  const int b = blockIdx.x >> 8;
  const int chunk = blockIdx.x & 255;
  const int n = chunk * 256 + tid;

  __builtin_prefetch(bboxes + ((size_t)b * Nn + n) * 4, 0, 3); // global_prefetch_b8

  if (tid < Mn) {
    async_copy_b128(&sgt[tid * 4], gtb + ((size_t)b * Mn + tid) * 4);
    sgm[tid] = __uint_as_float(gtmax[b * Mn + tid]);
    wait_async0();
  }
  __syncthreads();
  if (tid < Mn) {
    float x1 = sgt[tid*4], y1 = sgt[tid*4+1], x2 = sgt[tid*4+2], y2 = sgt[tid*4+3];
    sarea[tid] = (x2 - x1) * (y2 - y1);
  }
  __syncthreads();

  const float4 bb = reinterpret_cast<const float4*>(bboxes)[(size_t)b * Nn + n];
  const float ab = (bb.z - bb.x) * (bb.w - bb.y);
  float best = -1.0f;
  int arg = 0;
  int lq = -1;
  #pragma unroll 4
  for (int m = 0; m < Mn; ++m) {
    float lx = fmaxf(bb.x, sgt[m*4+0]);
    float ly = fmaxf(bb.y, sgt[m*4+1]);
    float rx = fminf(bb.z, sgt[m*4+2]);
    float ry = fminf(bb.w, sgt[m*4+3]);
    float w = fmaxf(rx - lx, 0.0f);
    float h = fmaxf(ry - ly, 0.0f);
    float inter = w * h;
    float iou = inter / (ab + sarea[m] - inter + EPSI);
    if (iou > best) { best = iou; arg = m; }          // first-max semantics
    if (lq < 0 && sgm[m] >= MIN_POS && iou == sgm[m]) lq = m; // bitwise-equal recompute
  }
  int a = (best >= POS_THR) ? (arg + 1) : ((best < NEG_THR) ? 0 : -1);
  if (lq >= 0) a = lq + 1;
  out[OUT_ASSIGNED + (size_t)b * Nn + n] = (float)a;
}

// ---- kernel 3: deterministic exact top-k (radix select + stable compaction
//                + bitonic sort desc with index tiebreak) ---------------------
__global__ __launch_bounds__(1024)
void select_kernel(const float* __restrict__ prio,
                   const float* __restrict__ out_assigned,
                   int* __restrict__ sel) {
  const int tid = threadIdx.x;
  const int b = blockIdx.x >> 1;
  const int isneg = blockIdx.x & 1;
  const int kSel = isneg ? NUM_NEG : NUM_POS;
  const float* pr = prio + (size_t)b * Nn;
  const float* as = out_assigned + (size_t)b * Nn;

  __shared__ unsigned hist[256];
  __shared__ unsigned sScan[1024];
  __shared__ float selKey[NUM_SAMPLES];
  __shared__ int selIdx[NUM_SAMPLES];
  __shared__ unsigned shPrefix, shPmask, shKk, shGtot;

  auto keyf = [&](int n) -> float {
    int a = (int)as[n];
    bool m = isneg ? (a == 0) : (a > 0);
    return pr[n] + (m ? BIGF : 0.0f);      // keys >= 0 -> uint-bit order == float order
  };

  if (tid == 0) { shPrefix = 0u; shPmask = 0u; shKk = (unsigned)kSel; }
  __syncthreads();

  // 4-pass MSB-first radix select for the kSel-th largest key
  for (int pass = 0; pass < 4; ++pass) {
    const int shift = 24 - 8 * pass;
    if (tid < 256) hist[tid] = 0u;
    __syncthreads();
    const unsigned prefix = shPrefix, pmask = shPmask;
    for (int n = tid; n < Nn; n += 1024) {
      unsigned u = __float_as_uint(keyf(n));
      if ((u & pmask) == prefix) atomicAdd(&hist[(u >> shift) & 255u], 1u);
    }
    __syncthreads();
    if (tid == 0) {
      unsigned kk = shKk, run = 0;
      int d = 255;
      for (; d > 0; --d) {
        if (run + hist[d] >= kk) break;
        run += hist[d];
      }
      shKk = kk - run;
      shPrefix = prefix | ((unsigned)d << shift);
      shPmask = pmask | (255u << shift);
    }
    __syncthreads();
  }
  const unsigned T = shPrefix; // threshold bits; need (kSel - count(>T)) elems == T

  // stable, atomic-free compaction: each thread owns 64 consecutive anchors
  const int CH = Nn / 1024;
  const int base = tid * CH;
  unsigned g = 0, e = 0;
  for (int j = 0; j < CH; ++j) {
    unsigned u = __float_as_uint(keyf(base + j));
    g += (u > T);
    e += (u == T);
  }
  // inclusive scan of g (Hillis-Steele)
  sScan[tid] = g; __syncthreads();
  for (int off = 1; off < 1024; off <<= 1) {
    unsigned v = sScan[tid];
    if (tid >= off) v += sScan[tid - off];
    __syncthreads(); sScan[tid] = v; __syncthreads();
  }
  const unsigned exclG = sScan[tid] - g;
  if (tid == 0) shGtot = sScan[1023];
  __syncthreads();
  const unsigned Gtot = shGtot;
  // inclusive scan of e
  sScan[tid] = e; __syncthreads();
  for (int off = 1; off < 1024; off <<= 1) {
    unsigned v = sScan[tid];
    if (tid >= off) v += sScan[tid - off];
    __syncthreads(); sScan[tid] = v; __syncthreads();
  }
  const unsigned exclE = sScan[tid] - e;
  __syncthreads();

  unsigned wG = exclG, wE = Gtot + exclE;
  for (int j = 0; j < CH; ++j) {
    const int n = base + j;
    const float kv = keyf(n);
    const unsigned u = __float_as_uint(kv);
    if (u > T) { selKey[wG] = kv; selIdx[wG] = n; ++wG; }
    else if (u == T) {
      if (wE < (unsigned)kSel) { selKey[wE] = kv; selIdx[wE] = n; }
      ++wE;
    }
  }
  __syncthreads();
  if (tid < NUM_SAMPLES && tid >= kSel) { selKey[tid] = -1.0f; selIdx[tid] = Nn; }
  __syncthreads();

  // bitonic sort: key descending, index ascending on ties (top_k semantics)
  for (unsigned ksz = 2; ksz <= NUM_SAMPLES; ksz <<= 1) {
    for (unsigned j = ksz >> 1; j > 0; j >>= 1) {
      if (tid < NUM_SAMPLES) {
        const unsigned ixj = (unsigned)tid ^ j;
        if (ixj > (unsigned)tid) {
          float k1 = selKey[tid], k2 = selKey[ixj];
          int i1 = selIdx[tid], i2 = selIdx[ixj];
          bool before = (k1 > k2) || (k1 == k2 && i1 < i2); // desc ordering
          bool ascBlock = (tid & ksz) != 0;
          bool doswap = ascBlock ? before : !before;
          if (doswap) {
            selKey[tid] = k2; selKey[ixj] = k1;
            selIdx[tid] = i2; selIdx[ixj] = i1;
          }
        }
      }
      __syncthreads();
    }
  }
  if (tid < kSel)
    sel[b * NUM_SAMPLES + (isneg ? NUM_POS : 0) + tid] = selIdx[tid];
}

// ---- kernel 4: gather rois / targets / labels / valid -----------------------
__global__ __launch_bounds__(256)
void gather_kernel(const float* __restrict__ bboxes,
                   const float* __restrict__ gtb,
                   const int* __restrict__ labels,
                   const int* __restrict__ sel,
                   float* __restrict__ out) {
  const int t = blockIdx.x * blockDim.x + threadIdx.x;
  if (t >= Bn * NUM_SAMPLES) return;
  const int b = t >> 9;
  const int i = t & 511;
  const int idx = sel[t];
  const int a = (int)out[OUT_ASSIGNED + (size_t)b * Nn + idx];
  const float4 bb = reinterpret_cast<const float4*>(bboxes)[(size_t)b * Nn + idx];
  float r[8];
  r[0] = bb.x; r[1] = bb.y; r[2] = bb.z; r[3] = bb.w;
  float lab = 0.0f, val = 0.0f;
  if (i < NUM_POS) {
    const bool pv = a > 0;
    int g = a - 1; if (g < 0) g = 0;
    const float4 gb = reinterpret_cast<const float4*>(gtb)[b * Mn + g];
    r[4] = pv ? gb.x : 0.0f; r[5] = pv ? gb.y : 0.0f;
    r[6] = pv ? gb.z : 0.0f; r[7] = pv ? gb.w : 0.0f;
    lab = pv ? (float)labels[b * Mn + g] : -1.0f;
    val = pv ? 1.0f : 0.0f;
  } else {
    r[4] = r[5] = r[6] = r[7] = 0.0f;
    val = (a == 0) ? 1.0f : 0.0f;
    lab = 0.0f;
  }
  float* ro = out + OUT_ROIS + (size_t)t * 8;
  #pragma unroll
  for (int q = 0; q < 8; ++q) ro[q] = r[q];
  out[OUT_LABELS + t] = lab;
  out[OUT_VALID + t] = val;
}

// ---- host launcher ----------------------------------------------------------
extern "C" void kernel_launch(void* const* d_in, const int* in_sizes, int n_in,
                              void* d_out, int out_size, void* d_ws, size_t ws_size,
                              hipStream_t stream) {
  const float* bboxes = (const float*)d_in[0];   // [B,N,4] f32
  const float* gtb    = (const float*)d_in[1];   // [B,M,4] f32
  const int*   labels = (const int*)d_in[2];     // [B,M] i32
  const float* prio   = (const float*)d_in[3];   // [B,N] f32
  float* out = (float*)d_out;

  unsigned* gtmax = (unsigned*)d_ws;                              // B*M uints
  int* sel = (int*)((char*)d_ws + Bn * Mn * sizeof(unsigned));    // B*512 ints

  init_gtmax_kernel<<<(Bn * Mn + 255) / 256, 256, 0, stream>>>(gtmax);
  iou_gtmax_kernel<<<Bn * (Nn / 256), 256, 0, stream>>>(bboxes, gtb, gtmax, out);
  assign_kernel<<<Bn * (Nn / 256), 256, 0, stream>>>(bboxes, gtb, gtmax, out);
  select_kernel<<<Bn * 2, 1024, 0, stream>>>(prio, out + OUT_ASSIGNED, sel);
  gather_kernel<<<(Bn * NUM_SAMPLES + 255) / 256, 256, 0, stream>>>(
      bboxes, gtb, labels, sel, out);
}